// GCNLayer_31628139168304
// MI455X (gfx1250) — compile-verified
//
#include <hip/hip_runtime.h>

#define D_FEAT 64
#define SLOPE 0.2f
#define TILE 256

typedef float v2f __attribute__((ext_vector_type(2)));
typedef float v8f __attribute__((ext_vector_type(8)));

// ---------------- CDNA5 async global->LDS (probe-verified on this toolchain) ----------------
#if defined(__AMDGCN__) && __has_builtin(__builtin_amdgcn_global_load_async_to_lds_b32)
#define HAVE_ASYNC_LDS 1
typedef __attribute__((address_space(1))) int* as1ip;
typedef __attribute__((address_space(3))) int* as3ip;
__device__ __forceinline__ void async_copy_b32(const void* g, void* l) {
  __builtin_amdgcn_global_load_async_to_lds_b32((as1ip)g, (as3ip)l, 0, 0);
}
#else
#define HAVE_ASYNC_LDS 0
#endif

__device__ __forceinline__ void wait_async_lds() {
#if HAVE_ASYNC_LDS
#if __has_builtin(__builtin_amdgcn_s_wait_asynccnt)
  __builtin_amdgcn_s_wait_asynccnt(0);
#else
  asm volatile("s_wait_asynccnt 0x0" ::: "memory");
#endif
#endif
}

// ---------------- CDNA5 WMMA probe ----------------
#if defined(__AMDGCN__) && __has_builtin(__builtin_amdgcn_wmma_f32_16x16x4_f32)
#define HAVE_WMMA_F32 1
#else
#define HAVE_WMMA_F32 0
#endif

__device__ __forceinline__ int rl(int x) { return __builtin_amdgcn_readfirstlane(x); }
__device__ __forceinline__ float rlf(float x) {
  return __int_as_float(__builtin_amdgcn_readfirstlane(__float_as_int(x)));
}

// ---------------- CSR-build kernels ----------------
__global__ void zero_i32_kernel(int* __restrict__ p, int n) {
  int i = blockIdx.x * blockDim.x + threadIdx.x;
  if (i < n) p[i] = 0;
}

__global__ void hist_kernel(const int* __restrict__ rows, int* __restrict__ counts, int E) {
  int i = blockIdx.x * blockDim.x + threadIdx.x;
  if (i < E) atomicAdd(&counts[rows[i]], 1);
}

// Single-workgroup exclusive scan over row counts (chunked Hillis-Steele).
__global__ void scan_kernel(const int* __restrict__ counts, int* __restrict__ offsets,
                            int* __restrict__ cursor, int n) {
  __shared__ int sums[1024];
  int t = threadIdx.x;
  int chunk = (n + 1023) / 1024;
  int beg = t * chunk;
  int end = (beg + chunk < n) ? beg + chunk : n;
  int s = 0;
  for (int i = beg; i < end; ++i) s += counts[i];
  sums[t] = s;
  __syncthreads();
  for (int off = 1; off < 1024; off <<= 1) {
    int v = sums[t];
    int add = (t >= off) ? sums[t - off] : 0;
    __syncthreads();
    sums[t] = v + add;
    __syncthreads();
  }
  int run = (t == 0) ? 0 : sums[t - 1];
  for (int i = beg; i < end; ++i) {
    offsets[i] = run;
    cursor[i] = run;
    run += counts[i];
  }
  if (t == 0) offsets[n] = sums[1023];
}

// Scatter edges into row-sorted order; edge stream double-buffered through LDS
// via GLOBAL_LOAD_ASYNC_TO_LDS_B32 so next tile's loads overlap the atomics.
__global__ void scatter_kernel(const int* __restrict__ rows, const int* __restrict__ cols,
                               const float* __restrict__ vals, int* __restrict__ cursor,
                               int* __restrict__ scol, float* __restrict__ sval, int E) {
  __shared__ int s_row[2][TILE];
  __shared__ int s_col[2][TILE];
  __shared__ float s_val[2][TILE];
  int t = threadIdx.x;
  int ntiles = (E + TILE - 1) / TILE;
  int tile = blockIdx.x;
  int buf = 0;
  if (tile < ntiles) {
    int e = tile * TILE + t;
    if (e < E) {
#if HAVE_ASYNC_LDS
      async_copy_b32(rows + e, &s_row[buf][t]);
      async_copy_b32(cols + e, &s_col[buf][t]);
      async_copy_b32(vals + e, &s_val[buf][t]);
#else
      s_row[buf][t] = rows[e]; s_col[buf][t] = cols[e]; s_val[buf][t] = vals[e];
#endif
    }
  }
  for (; tile < ntiles; tile += gridDim.x) {
    wait_async_lds();
    __syncthreads();
    int next = tile + gridDim.x;
    if (next < ntiles) {
      int nb = buf ^ 1;
      int e = next * TILE + t;
      if (e < E) {
#if HAVE_ASYNC_LDS
        async_copy_b32(rows + e, &s_row[nb][t]);
        async_copy_b32(cols + e, &s_col[nb][t]);
        async_copy_b32(vals + e, &s_val[nb][t]);
#else
        s_row[nb][t] = rows[e]; s_col[nb][t] = cols[e]; s_val[nb][t] = vals[e];
#endif
      }
    }
    int e = tile * TILE + t;
    if (e < E) {
      int r = s_row[buf][t];
      int pos = atomicAdd(&cursor[r], 1);
      scol[pos] = s_col[buf][t];
      sval[pos] = s_val[buf][t];
    }
    buf ^= 1;
  }
}

// One wave32 per output row. 16-edge chunks of the row reduction are computed
// on the matrix pipe: D(16x16) += A(16x4) x B(4x16) chained 4x over K, where A
// broadcasts the 4 edge vals across M (per the ISA A-layout: VGPR0=K0/K2 per
// lane-half, VGPR1=K1/K3) and B holds the 4 gathered feature rows. Every row
// of D then equals the row-reduction, so D-VGPR0 directly holds out[c*16+n]
// replicated across lane halves. Tail (<16 edges) uses scalar FMAs in the same
// per-lane column ownership (lo half: cols 0-15/16-31, hi half: 32-47/48-63).
__global__ void spmm_rows_kernel(const int* __restrict__ offsets, const int* __restrict__ scol,
                                 const float* __restrict__ sval, const float* __restrict__ embeds,
                                 float* __restrict__ out, int N) {
  int wave = (blockIdx.x * blockDim.x + threadIdx.x) >> 5;
  int lane = threadIdx.x & 31;
  if (wave >= N) return;
  int r = rl(wave);
  int beg = rl(offsets[r]);
  int end = rl(offsets[r + 1]);
  int nlo = lane & 15;
  bool hi = lane >= 16;
  int colA = (hi ? 32 : 0) + nlo;   // this lane's first owned feature column
  int colB = (hi ? 48 : 16) + nlo;  // this lane's second owned feature column
  float accA = 0.0f, accB = 0.0f;
  int e = beg;

#if HAVE_WMMA_F32
  v8f d0 = {}, d1 = {}, d2 = {}, d3 = {};
  for (; e + 16 <= end; e += 16) {
    if (e + 16 < end) {  // global_prefetch upcoming gather rows toward L2/L0
      __builtin_prefetch(embeds + (size_t)rl(scol[e + 16]) * D_FEAT, 0, 0);
    }
#pragma unroll
    for (int g = 0; g < 4; ++g) {
      int eg = e + g * 4;
      int c0 = rl(scol[eg + 0]), c1 = rl(scol[eg + 1]);
      int c2 = rl(scol[eg + 2]), c3 = rl(scol[eg + 3]);
      float v0 = rlf(sval[eg + 0]), v1 = rlf(sval[eg + 1]);
      float v2 = rlf(sval[eg + 2]), v3 = rlf(sval[eg + 3]);
      // A 16x4 f32: VGPR0 = {K0 (lanes 0-15), K2 (lanes 16-31)}, VGPR1 = {K1, K3};
      // vals broadcast across M so every D row computes the same reduction.
      v2f A;
      A.x = hi ? v2 : v0;
      A.y = hi ? v3 : v1;
      // B 4x16 f32 per 16-col chunk, K-mapping mirroring A's per-half layout.
      const float* pKa = embeds + (size_t)(hi ? c2 : c0) * D_FEAT + nlo;  // K0 / K2
      const float* pKb = embeds + (size_t)(hi ? c3 : c1) * D_FEAT + nlo;  // K1 / K3
      v2f B0, B1, B2, B3;
      B0.x = pKa[0];  B0.y = pKb[0];
      B1.x = pKa[16]; B1.y = pKb[16];
      B2.x = pKa[32]; B2.y = pKb[32];
      B3.x = pKa[48]; B3.y = pKb[48];
      d0 = __builtin_amdgcn_wmma_f32_16x16x4_f32(false, A, false, B0, (short)0, d0, false, false);
      d1 = __builtin_amdgcn_wmma_f32_16x16x4_f32(false, A, false, B1, (short)0, d1, false, false);
      d2 = __builtin_amdgcn_wmma_f32_16x16x4_f32(false, A, false, B2, (short)0, d2, false, false);
      d3 = __builtin_amdgcn_wmma_f32_16x16x4_f32(false, A, false, B3, (short)0, d3, false, false);
    }
  }
  // D-VGPR0 of each chunk holds out[c*16 + nlo], valid in both lane halves.
  accA = hi ? d2[0] : d0[0];
  accB = hi ? d3[0] : d1[0];
#endif

  for (; e < end; ++e) {  // scalar tail (and full path when WMMA unavailable)
    int c = rl(scol[e]);
    float v = rlf(sval[e]);
    if (e + 2 < end) {
      __builtin_prefetch(embeds + (size_t)rl(scol[e + 2]) * D_FEAT, 0, 0);
    }
    const float* p = embeds + (size_t)c * D_FEAT;
    accA = fmaf(v, p[colA], accA);
    accB = fmaf(v, p[colB], accB);
  }

  accA = (accA >= 0.0f) ? accA : SLOPE * accA;  // fused LeakyReLU
  accB = (accB >= 0.0f) ? accB : SLOPE * accB;
  out[(size_t)r * D_FEAT + colA] = accA;
  out[(size_t)r * D_FEAT + colB] = accB;
}

// ---------------- fallback path (if workspace too small for CSR) ----------------
__global__ void zero_f32_kernel(float* __restrict__ p, int n) {
  int i = blockIdx.x * blockDim.x + threadIdx.x;
  if (i < n) p[i] = 0.0f;
}

__global__ void edge_atomic_kernel(const int* __restrict__ rows, const int* __restrict__ cols,
                                   const float* __restrict__ vals,
                                   const float* __restrict__ embeds, float* __restrict__ out,
                                   int E) {
  int wave = (blockIdx.x * blockDim.x + threadIdx.x) >> 5;
  int lane = threadIdx.x & 31;
  if (wave >= E) return;
  int e = rl(wave);
  int r = rl(rows[e]);
  int c = rl(cols[e]);
  float v = rlf(vals[e]);
  float2 f = ((const float2*)embeds)[(size_t)c * (D_FEAT / 2) + lane];
  atomicAdd(out + (size_t)r * D_FEAT + lane * 2, v * f.x);
  atomicAdd(out + (size_t)r * D_FEAT + lane * 2 + 1, v * f.y);
}

__global__ void leaky_kernel(float* __restrict__ out, int n) {
  int i = blockIdx.x * blockDim.x + threadIdx.x;
  if (i < n) {
    float x = out[i];
    out[i] = (x >= 0.0f) ? x : SLOPE * x;
  }
}

extern "C" void kernel_launch(void* const* d_in, const int* in_sizes, int n_in,
                              void* d_out, int out_size, void* d_ws, size_t ws_size,
                              hipStream_t stream) {
  const int* rows = (const int*)d_in[0];
  const int* cols = (const int*)d_in[1];
  const float* vals = (const float*)d_in[2];
  const float* embeds = (const float*)d_in[3];
  float* out = (float*)d_out;
  const int E = in_sizes[0];
  const int N = in_sizes[3] / D_FEAT;

  // CSR scratch: counts[N] | offsets[N+1] | cursor[N] | scol[E] | sval[E]
  size_t need = ((size_t)3 * N + 1 + 2 * (size_t)E) * sizeof(int);
  if (ws_size >= need) {
    int* counts = (int*)d_ws;
    int* offsets = counts + N;
    int* cursor = offsets + N + 1;
    int* scol = cursor + N;
    float* sval = (float*)(scol + E);

    zero_i32_kernel<<<(N + 255) / 256, 256, 0, stream>>>(counts, N);
    hist_kernel<<<(E + 255) / 256, 256, 0, stream>>>(rows, counts, E);
    scan_kernel<<<1, 1024, 0, stream>>>(counts, offsets, cursor, N);
    int ntiles = (E + TILE - 1) / TILE;
    int sblocks = (ntiles < 1024) ? ntiles : 1024;
    scatter_kernel<<<sblocks, TILE, 0, stream>>>(rows, cols, vals, cursor, scol, sval, E);
    spmm_rows_kernel<<<(N + 7) / 8, 256, 0, stream>>>(offsets, scol, sval, embeds, out, N);
  } else {
    zero_f32_kernel<<<(N * D_FEAT + 255) / 256, 256, 0, stream>>>(out, N * D_FEAT);
    edge_atomic_kernel<<<(E + 7) / 8, 256, 0, stream>>>(rows, cols, vals, embeds, out, E);
    leaky_kernel<<<(N * D_FEAT + 255) / 256, 256, 0, stream>>>(out, N * D_FEAT);
  }
}